// CircuitTrainingModel_58145267254193
// MI455X (gfx1250) — compile-verified
//
#include <hip/hip_runtime.h>

typedef __attribute__((ext_vector_type(2))) float v2f;
typedef __attribute__((ext_vector_type(8))) float v8f;

// ---------------------------------------------------------------------------
// Degree / normalization kernels
// ---------------------------------------------------------------------------
__global__ void init_deg_kernel(float* __restrict__ deg, int n) {
    int i = blockIdx.x * blockDim.x + threadIdx.x;
    if (i < n) deg[i] = 1.0f;  // self-loop contribution
}

__global__ void edge_deg_kernel(const int* __restrict__ dst, float* __restrict__ deg, int E) {
    int e = blockIdx.x * blockDim.x + threadIdx.x;
    if (e < E) atomicAdd(&deg[dst[e]], 1.0f);
}

__global__ void deg_to_dinv_kernel(float* __restrict__ deg, int n) {
    int i = blockIdx.x * blockDim.x + threadIdx.x;
    if (i < n) {
        float d = deg[i];
        deg[i] = (d > 0.0f) ? rsqrtf(d) : 0.0f;
    }
}

// Pad x [n x 6] -> xp [n x 8] (zero-fill cols 6,7).
__global__ void pad_x_kernel(const float* __restrict__ x, float* __restrict__ xp, int n) {
    int t = blockIdx.x * blockDim.x + threadIdx.x;
    if (t >= n * 8) return;
    int i = t >> 3, k = t & 7;
    xp[t] = (k < 6) ? x[i * 6 + k] : 0.0f;
}

// Pad W1 [6 x 64] -> Wp [8 x 64] (zero rows 6,7).
__global__ void pad_w_kernel(const float* __restrict__ w, float* __restrict__ wp) {
    int t = blockIdx.x * blockDim.x + threadIdx.x;  // 0..511
    if (t >= 8 * 64) return;
    wp[t] = (t < 6 * 64) ? w[t] : 0.0f;
}

// ---------------------------------------------------------------------------
// fp32 WMMA GEMM: Out[nrows x (NT*16)] = X[nrows x K] @ W[K x (NT*16)]
// One wave per 16-row strip; A fragment reused across NT column tiles.
// K, NCOLS compile-time -> unconditional b64/b32 loads, fully unrolled,
// (K/4)*NT static v_wmma_f32_16x16x4_f32 per wave.
// ---------------------------------------------------------------------------
template <int K, int NT, bool BIAS_RELU>
__global__ __launch_bounds__(32) void gemm_wmma(
    const float* __restrict__ X, const float* __restrict__ W,
    float* __restrict__ Out, const float* __restrict__ bias)
{
    constexpr int NCOLS = NT * 16;
    const int lane = threadIdx.x;   // wave32
    const int hi   = lane >> 4;     // 0 / 1
    const int lo   = lane & 15;
    const int row0 = blockIdx.x * 16;

    const float* Arow = X + (size_t)(row0 + lo) * K;

    v8f c[NT];
#pragma unroll
    for (int t = 0; t < NT; ++t) c[t] = (v8f){};

#pragma unroll
    for (int k0 = 0; k0 < K; k0 += 4) {
        // A 16x4 f32: VGPR0 = K=k0 (lanes 0-15) / k0+2 (lanes 16-31), VGPR1 -> +1
        const int ka = k0 + 2 * hi;
        const v2f a = *(const v2f*)(Arow + ka);          // one global_load_b64
#pragma unroll
        for (int t = 0; t < NT; ++t) {
            v2f b;
            b.x = W[ka * NCOLS + 16 * t + lo];           // row ka   of B tile t
            b.y = W[(ka + 1) * NCOLS + 16 * t + lo];     // row ka+1
            c[t] = __builtin_amdgcn_wmma_f32_16x16x4_f32(
                false, a, false, b, (short)0, c[t], false, false);
        }
    }

    // C/D 16x16 f32: VGPR j -> M=j (lanes 0-15) / M=j+8 (lanes 16-31), N=lane%16
#pragma unroll
    for (int t = 0; t < NT; ++t) {
        const int col = 16 * t + lo;
#pragma unroll
        for (int j = 0; j < 8; ++j) {
            float v = c[t][j];
            if (BIAS_RELU) v = fmaxf(v + bias[col], 0.0f);
            Out[(size_t)(row0 + j + 8 * hi) * NCOLS + col] = v;
        }
    }
}

// ---------------------------------------------------------------------------
// GCN aggregation
// ---------------------------------------------------------------------------
__global__ void self_init_kernel(const float* __restrict__ tmp,
                                 const float* __restrict__ dinv,
                                 float* __restrict__ agg, int n) {
    long long t = (long long)blockIdx.x * blockDim.x + threadIdx.x;
    if (t >= (long long)n * 64) return;
    int i = (int)(t >> 6);
    float di = dinv[i];
    agg[t] = tmp[t] * di * di;
}

__global__ void edge_msg_kernel(const int* __restrict__ src,
                                const int* __restrict__ dst,
                                const float* __restrict__ dinv,
                                const float* __restrict__ tmp,
                                float* __restrict__ agg, int E) {
    long long t = (long long)blockIdx.x * blockDim.x + threadIdx.x;
    int e    = (int)(t >> 5);
    int lane = (int)(t & 31);
    if (e >= E) return;
    int s = src[e], d = dst[e];
    float nrm = dinv[s] * dinv[d];
    const float2 v = *(const float2*)(tmp + (size_t)s * 64 + 2 * lane);
    atomicAdd(agg + (size_t)d * 64 + 2 * lane,     v.x * nrm);
    atomicAdd(agg + (size_t)d * 64 + 2 * lane + 1, v.y * nrm);
}

__global__ void bias_relu_kernel(float* __restrict__ h,
                                 const float* __restrict__ bias, int n) {
    long long t = (long long)blockIdx.x * blockDim.x + threadIdx.x;
    if (t >= (long long)n * 64) return;
    int f = (int)(t & 63);
    h[t] = fmaxf(h[t] + bias[f], 0.0f);
}

// ---------------------------------------------------------------------------
// Global mean pool
// ---------------------------------------------------------------------------
__global__ void pool_zero_kernel(float* __restrict__ sums, float* __restrict__ cnts,
                                 int ng, int hid) {
    int t = blockIdx.x * blockDim.x + threadIdx.x;
    if (t < ng * hid) sums[t] = 0.0f;
    if (t < ng)       cnts[t] = 0.0f;
}

__global__ void pool_add_kernel(const float* __restrict__ h,
                                const int* __restrict__ batch,
                                float* __restrict__ sums, int n) {
    long long t = (long long)blockIdx.x * blockDim.x + threadIdx.x;
    if (t >= (long long)n * 64) return;
    int i = (int)(t >> 6);
    int f = (int)(t & 63);
    atomicAdd(sums + (size_t)batch[i] * 64 + f, h[t]);
}

__global__ void pool_cnt_kernel(const int* __restrict__ batch,
                                float* __restrict__ cnts, int n) {
    int i = blockIdx.x * blockDim.x + threadIdx.x;
    if (i < n) atomicAdd(&cnts[batch[i]], 1.0f);
}

__global__ void pool_div_kernel(const float* __restrict__ sums,
                                const float* __restrict__ cnts,
                                float* __restrict__ G, int ng, int hid) {
    int t = blockIdx.x * blockDim.x + threadIdx.x;
    if (t >= ng * hid) return;
    int g = t / hid;
    G[t] = sums[t] / fmaxf(cnts[g], 1.0f);
}

// ---------------------------------------------------------------------------
// Host launcher
// ---------------------------------------------------------------------------
extern "C" void kernel_launch(void* const* d_in, const int* in_sizes, int n_in,
                              void* d_out, int out_size, void* d_ws, size_t ws_size,
                              hipStream_t stream) {
    (void)n_in; (void)out_size; (void)ws_size;

    const float* x     = (const float*)d_in[0];
    const int*   ei    = (const int*)  d_in[1];
    const int*   batch = (const int*)  d_in[2];
    const float* W1  = (const float*)d_in[3];
    const float* b1  = (const float*)d_in[4];
    const float* W2  = (const float*)d_in[5];
    const float* b2  = (const float*)d_in[6];
    const float* W3  = (const float*)d_in[7];
    const float* b3  = (const float*)d_in[8];
    const float* Wfc = (const float*)d_in[9];
    const float* bfc = (const float*)d_in[10];

    const int FIN = 6, HID = 64, EMB = 128, NG = 64;
    const int N = in_sizes[0] / FIN;
    const int E = in_sizes[1] / 2;
    const int* srcIdx = ei;       // edge_index row 0
    const int* dstIdx = ei + E;   // edge_index row 1

    // workspace carve-out (~28 MB)
    char* ws = (char*)d_ws;
    auto a256 = [](size_t v) { return (v + 255) & ~(size_t)255; };
    size_t off = 0;
    float* dinv = (float*)(ws + off); off += a256((size_t)N * 4);
    float* xpad = (float*)(ws + off); off += a256((size_t)N * 8 * 4);
    float* Wpad = (float*)(ws + off); off += a256((size_t)8 * HID * 4);
    float* bufA = (float*)(ws + off); off += a256((size_t)N * HID * 4);
    float* bufB = (float*)(ws + off); off += a256((size_t)N * HID * 4);
    float* sums = (float*)(ws + off); off += a256((size_t)NG * HID * 4);
    float* cnts = (float*)(ws + off); off += a256((size_t)NG * 4);
    float* G    = (float*)(ws + off); off += a256((size_t)NG * HID * 4);

    const int TB = 256;
    const long long nh = (long long)N * HID;
    const unsigned nhb = (unsigned)((nh + TB - 1) / TB);
    const unsigned eb  = (unsigned)(((long long)E * 32 + TB - 1) / TB);

    // symmetric normalization: dinv = rsqrt(deg), deg includes self-loop
    init_deg_kernel<<<(N + TB - 1) / TB, TB, 0, stream>>>(dinv, N);
    edge_deg_kernel<<<(E + TB - 1) / TB, TB, 0, stream>>>(dstIdx, dinv, E);
    deg_to_dinv_kernel<<<(N + TB - 1) / TB, TB, 0, stream>>>(dinv, N);

    // pad x [Nx6]->[Nx8], W1 [6x64]->[8x64]
    pad_x_kernel<<<(N * 8 + TB - 1) / TB, TB, 0, stream>>>(x, xpad, N);
    pad_w_kernel<<<2, TB, 0, stream>>>(W1, Wpad);

    const int rowTiles = N / 16;  // 50000 / 16 = 3125 exact

    // ---- layer 1 (K=8 padded) ----
    gemm_wmma<8, 4, false><<<rowTiles, 32, 0, stream>>>(xpad, Wpad, bufA, nullptr);
    self_init_kernel<<<nhb, TB, 0, stream>>>(bufA, dinv, bufB, N);
    edge_msg_kernel<<<eb, TB, 0, stream>>>(srcIdx, dstIdx, dinv, bufA, bufB, E);
    bias_relu_kernel<<<nhb, TB, 0, stream>>>(bufB, b1, N);

    // ---- layers 2,3 (K=64) ----
    const float* Wl[2] = {W2, W3};
    const float* bl[2] = {b2, b3};
    for (int l = 0; l < 2; ++l) {
        gemm_wmma<64, 4, false><<<rowTiles, 32, 0, stream>>>(bufB, Wl[l], bufA, nullptr);
        self_init_kernel<<<nhb, TB, 0, stream>>>(bufA, dinv, bufB, N);
        edge_msg_kernel<<<eb, TB, 0, stream>>>(srcIdx, dstIdx, dinv, bufA, bufB, E);
        bias_relu_kernel<<<nhb, TB, 0, stream>>>(bufB, bl[l], N);
    }

    // global mean pool
    pool_zero_kernel<<<(NG * HID + TB - 1) / TB, TB, 0, stream>>>(sums, cnts, NG, HID);
    pool_add_kernel<<<nhb, TB, 0, stream>>>(bufB, batch, sums, N);
    pool_cnt_kernel<<<(N + TB - 1) / TB, TB, 0, stream>>>(batch, cnts, N);
    pool_div_kernel<<<(NG * HID + TB - 1) / TB, TB, 0, stream>>>(sums, cnts, G, NG, HID);

    // FC: [64 x 64] @ [64 x 128] + bias, relu -> d_out
    gemm_wmma<64, 8, true><<<NG / 16, 32, 0, stream>>>(G, Wfc, (float*)d_out, bfc);
}